// CenterLoss_29094108463751
// MI455X (gfx1250) — compile-verified
//
#include <hip/hip_runtime.h>
#include <hip/hip_bf16.h>

// CenterLoss on MI455X (gfx1250, wave32).
// loss = (1/B) * sum_b ||x_b - centers[labels_b]||^2
// Memory-bound (~16 MB traffic -> ~0.7 us at 23.3 TB/s). The 512-dim row
// reduction is mapped onto V_WMMA_F32_16X16X4_F32 (f32 in/out, matches
// reference precision): A = squared diffs in the ISA 16x4 f32 layout,
// B = ones, D accumulates row sums.

typedef __attribute__((ext_vector_type(2))) float v2f;
typedef __attribute__((ext_vector_type(8))) float v8f;

__global__ void __launch_bounds__(256)
center_dist_kernel(const float* __restrict__ x,
                   const int* __restrict__ labels,
                   const float* __restrict__ centers,
                   float* __restrict__ row_dist,
                   int feat) {
  const int lane        = threadIdx.x & 31;
  const int waveInBlock = threadIdx.x >> 5;
  const int waveId      = blockIdx.x * (blockDim.x >> 5) + waveInBlock;
  const int rowBase     = waveId * 16;

  // ISA 32-bit A-matrix 16x4 layout:
  //   lanes 0-15  : M = lane,    VGPR0 = K0, VGPR1 = K1
  //   lanes 16-31 : M = lane-16, VGPR0 = K2, VGPR1 = K3
  const int m     = lane & 15;            // row within 16-row tile
  const int khalf = (lane >> 4) * 2;      // K offset: 0 or 2
  const int row   = rowBase + m;
  const int lab   = labels[row];

  const float* xp = x       + (size_t)row * feat + khalf;
  const float* cp = centers + (size_t)lab * feat + khalf;

  v8f acc = {};                            // C/D accumulator (16x16 f32)
  v2f ones; ones.x = 1.0f; ones.y = 1.0f;  // B = all ones -> D[m,n] = rowsum(m)

  const int chunks = feat >> 2;            // K = 4 per WMMA
#pragma unroll 4
  for (int k = 0; k < chunks; ++k) {
    v2f xv = *(const v2f*)xp;              // global_load_b64, 8B aligned
    v2f cv = *(const v2f*)cp;
    xp += 4; cp += 4;
    v2f d; d.x = xv.x - cv.x; d.y = xv.y - cv.y;
    v2f a; a.x = d.x * d.x;   a.y = d.y * d.y;
    // D = A * ones + C  (all lanes active -> EXEC all 1s as required)
    acc = __builtin_amdgcn_wmma_f32_16x16x4_f32(
        /*neg_a=*/false, a, /*neg_b=*/false, ones,
        /*c_mod=*/(short)0, acc, /*reuse_a=*/false, /*reuse_b=*/false);
  }

  // C/D layout: lanes 0-15 VGPR v -> M=v (col = lane); lanes 16-31 -> M=8+v.
  // All columns are identical (B = ones), so lane 0 / lane 16 suffice.
  if (lane == 0) {
#pragma unroll
    for (int v = 0; v < 8; ++v) row_dist[rowBase + v] = acc[v];
  } else if (lane == 16) {
#pragma unroll
    for (int v = 0; v < 8; ++v) row_dist[rowBase + 8 + v] = acc[v];
  }
}

// Deterministic fixed-order reduction of the B per-row distances.
__global__ void __launch_bounds__(256)
reduce_mean_kernel(const float* __restrict__ row_dist,
                   float* __restrict__ out, int n) {
  __shared__ float s[256];
  float acc = 0.0f;
  for (int i = threadIdx.x; i < n; i += 256) acc += row_dist[i];
  s[threadIdx.x] = acc;
  __syncthreads();
#pragma unroll
  for (int off = 128; off > 0; off >>= 1) {
    if ((int)threadIdx.x < off) s[threadIdx.x] += s[threadIdx.x + off];
    __syncthreads();
  }
  if (threadIdx.x == 0) out[0] = s[0] / (float)n;
}

extern "C" void kernel_launch(void* const* d_in, const int* in_sizes, int n_in,
                              void* d_out, int out_size, void* d_ws, size_t ws_size,
                              hipStream_t stream) {
  const float* x       = (const float*)d_in[0];
  const int*   labels  = (const int*)d_in[1];
  const float* centers = (const float*)d_in[2];
  float*       out     = (float*)d_out;
  float*       row_d   = (float*)d_ws;     // B floats of scratch

  const int B = in_sizes[1];               // 4096 (labels count)
  const int F = in_sizes[0] / B;           // 512

  const int wavesPerBlock = 8;             // 256 threads = 8 wave32
  const int waves  = B / 16;               // one wave per 16-row tile
  const int blocks = waves / wavesPerBlock;

  center_dist_kernel<<<blocks, wavesPerBlock * 32, 0, stream>>>(
      x, labels, centers, row_d, F);
  reduce_mean_kernel<<<1, 256, 0, stream>>>(row_d, out, B);
}